// FocalLoss_2886218023270
// MI455X (gfx1250) — compile-verified
//
#include <hip/hip_runtime.h>
#include <stdint.h>

// RetinaNet focal loss for MI455X (gfx1250, wave32).
// Bandwidth-bound: 307MB classification stream dominates (~13us @ 23.3TB/s).
// Wave reductions use V_WMMA_F32_16X16X4_F32 (A = lane partials as 16x4 tile,
// B = ones => D[m][n] = s_m + s_{m+16}; 8 adds + shfl_xor(16) completes sum).

#define FL_ALPHA 0.25f

typedef __attribute__((ext_vector_type(2))) float v2f;
typedef __attribute__((ext_vector_type(8))) float v8f;

__device__ __forceinline__ float wave_sum32(float v) {
#if __has_builtin(__builtin_amdgcn_wmma_f32_16x16x4_f32)
    // A (16x4 f32): lanes 0-15 {VGPR0=K0, VGPR1=K1}; lanes 16-31 {K2, K3}.
    // Put partial in K0/K2 slot, zero the other -> A[m][0]=s_m, A[m][2]=s_{m+16}.
    v2f a; a[0] = v; a[1] = 0.0f;
    v2f bm; bm[0] = 1.0f; bm[1] = 1.0f;       // B = all ones (layout-invariant)
    v8f c = {};
    v8f d = __builtin_amdgcn_wmma_f32_16x16x4_f32(
        false, a, false, bm, (short)0, c, false, false);
    // lane in 0..15 holds rows 0..7 of D, lanes 16..31 hold rows 8..15.
    float s = ((d[0] + d[1]) + (d[2] + d[3])) + ((d[4] + d[5]) + (d[6] + d[7]));
    s += __shfl_xor(s, 16, 32);               // combine the two row groups
    return s;
#else
    for (int off = 16; off > 0; off >>= 1) v += __shfl_xor(v, off, 32);
    return v;
#endif
}

__global__ void init_acc_kernel(float* acc, int n) {
    int i = blockIdx.x * blockDim.x + threadIdx.x;
    if (i < n) acc[i] = 0.0f;
}

// One block = 320 anchors of one image. Computes IoU assignment, packed state,
// regression smooth-L1 partials, and positive counts.
__global__ void assign_kernel(const float* __restrict__ anchors,
                              const float* __restrict__ regressions,
                              const float* __restrict__ annotations,
                              uint32_t* __restrict__ state,
                              float* __restrict__ acc,   // [cls B][reg B][npos B]
                              int N, int M, int B) {
    const int b = blockIdx.y;
    const int n = blockIdx.x * blockDim.x + threadIdx.x;
    const bool live = (n < N);
    const int nc = live ? n : (N - 1);

    __shared__ float sAnn[5 * 64];
    const int annCount = 5 * M;
    for (int i = threadIdx.x; i < annCount; i += blockDim.x)
        sAnn[i] = annotations[(size_t)b * annCount + i];
    __syncthreads();

    const float4 a = *(const float4*)(anchors + 4 * (size_t)nc);
    const float aw = a.z - a.x, ah = a.w - a.y;
    const float acx = a.x + 0.5f * aw, acy = a.y + 0.5f * ah;
    const float aarea = aw * ah;

    float best = -2.0f;       // below the -1.0 sentinel -> first-argmax semantics
    int   bidx = 0;
    int   nvalid = 0;
    for (int j = 0; j < M; ++j) {
        const float gx0 = sAnn[5 * j + 0], gy0 = sAnn[5 * j + 1];
        const float gx1 = sAnn[5 * j + 2], gy1 = sAnn[5 * j + 3];
        const bool valid = (sAnn[5 * j + 4] != -1.0f);
        nvalid += valid ? 1 : 0;
        const float iw = fminf(a.z, gx1) - fmaxf(a.x, gx0);
        const float ih = fminf(a.w, gy1) - fmaxf(a.y, gy0);
        const float inter = fmaxf(iw, 0.0f) * fmaxf(ih, 0.0f);
        const float uni = fmaxf(aarea + (gx1 - gx0) * (gy1 - gy0) - inter, 1e-8f);
        const float iou = valid ? (inter / uni) : -1.0f;
        if (iou > best) { best = iou; bidx = j; }
    }

    // type: 0 = negative, 1 = positive, 2 = ignore (zero cls loss)
    uint32_t type;
    if (nvalid == 0)          type = 2u;   // any_valid==False -> everything zero
    else if (best >= 0.5f)    type = 1u;
    else if (best <  0.4f)    type = 0u;
    else                      type = 2u;

    const int lbl = (int)sAnn[5 * bidx + 4];

    float regloss = 0.0f;
    float poscnt  = 0.0f;
    if (live && type == 1u) {
        poscnt = 1.0f;
        const float g0 = sAnn[5 * bidx + 0], g1 = sAnn[5 * bidx + 1];
        const float g2 = sAnn[5 * bidx + 2], g3 = sAnn[5 * bidx + 3];
        const float gw0 = g2 - g0, gh0 = g3 - g1;
        const float gcx = g0 + 0.5f * gw0, gcy = g1 + 0.5f * gh0;
        const float gw = fmaxf(gw0, 1.0f), gh = fmaxf(gh0, 1.0f);
        const float t0 = ((gcx - acx) / aw) / 0.1f;
        const float t1 = ((gcy - acy) / ah) / 0.1f;
        const float t2 = __logf(gw / aw) / 0.2f;
        const float t3 = __logf(gh / ah) / 0.2f;
        const float4 r = *(const float4*)(regressions + 4 * ((size_t)b * N + nc));
        const float d0 = fabsf(t0 - r.x), d1 = fabsf(t1 - r.y);
        const float d2 = fabsf(t2 - r.z), d3 = fabsf(t3 - r.w);
        const float th = 1.0f / 9.0f, hc = 0.5f / 9.0f;
        regloss  = (d0 <= th) ? 4.5f * d0 * d0 : d0 - hc;
        regloss += (d1 <= th) ? 4.5f * d1 * d1 : d1 - hc;
        regloss += (d2 <= th) ? 4.5f * d2 * d2 : d2 - hc;
        regloss += (d3 <= th) ? 4.5f * d3 * d3 : d3 - hc;
    }

    if (live)
        state[(size_t)b * N + nc] = type | ((uint32_t)(lbl & 0xff) << 8);

    // block reduce (all threads converged here -> EXEC all-ones for WMMA)
    const float wr = wave_sum32(regloss);
    const float wp = wave_sum32(poscnt);
    __shared__ float sReg, sPos;
    if (threadIdx.x == 0) { sReg = 0.0f; sPos = 0.0f; }
    __syncthreads();
    if ((threadIdx.x & 31) == 0) { atomicAdd(&sReg, wr); atomicAdd(&sPos, wp); }
    __syncthreads();
    if (threadIdx.x == 0) {
        atomicAdd(&acc[B + b],     sReg);
        atomicAdd(&acc[2 * B + b], sPos);
    }
}

// Streams the [B,N,C] classification tensor; one float4 (4 classes) per thread.
// blockDim 256 * 4 = 1024 elements; N*C % 1024 == 0 so each block is one image.
__global__ void cls_loss_kernel(const float* __restrict__ cls,
                                const uint32_t* __restrict__ state,
                                float* __restrict__ acc,
                                int N, int C) {
    const int b = blockIdx.y;
    const size_t imgElems = (size_t)N * (size_t)C;
    size_t e = (((size_t)blockIdx.x * blockDim.x) + threadIdx.x) * 4;
    const bool live = (e + 3 < imgElems);
    if (!live) e = 0;
    const int n  = (int)(e / (unsigned)C);
    const int c0 = (int)(e % (unsigned)C);

    const uint32_t st   = state[(size_t)b * N + n];
    const uint32_t type = st & 3u;
    const int      lbl  = (int)((st >> 8) & 0xffu);

    float partial = 0.0f;
    if (live && type != 2u) {   // ignore-band anchors: skip the load entirely
        const float4 p4 = *(const float4*)(cls + (size_t)b * imgElems + e);
#pragma unroll
        for (int k = 0; k < 4; ++k) {
            float p = (&p4.x)[k];
            p = fminf(fmaxf(p, 1e-4f), 1.0f - 1e-4f);
            const bool ispos = (type == 1u) && ((c0 + k) == lbl);
            float l;
            if (ispos) { const float q = 1.0f - p; l = FL_ALPHA * q * q * (-__logf(p)); }
            else       { l = (1.0f - FL_ALPHA) * p * p * (-__logf(1.0f - p)); }
            partial += l;
        }
    }

    const float w = wave_sum32(partial);
    __shared__ float sAcc;
    if (threadIdx.x == 0) sAcc = 0.0f;
    __syncthreads();
    if ((threadIdx.x & 31) == 0) atomicAdd(&sAcc, w);
    __syncthreads();
    if (threadIdx.x == 0) atomicAdd(&acc[b], sAcc);
}

__global__ void finalize_kernel(const float* __restrict__ acc,
                                float* __restrict__ out, int B) {
    if (blockIdx.x == 0 && threadIdx.x == 0) {
        float cm = 0.0f, rm = 0.0f;
        for (int b = 0; b < B; ++b) {
            const float np = acc[2 * B + b];
            cm += acc[b] / fmaxf(np, 1.0f);
            rm += (np > 0.0f) ? (acc[B + b] / fmaxf(4.0f * np, 1.0f)) : 0.0f;
        }
        out[0] = cm / (float)B;
        out[1] = rm / (float)B;
    }
}

extern "C" void kernel_launch(void* const* d_in, const int* in_sizes, int n_in,
                              void* d_out, int out_size, void* d_ws, size_t ws_size,
                              hipStream_t stream) {
    const float* classifications = (const float*)d_in[0];  // [B,N,C]
    const float* regressions     = (const float*)d_in[1];  // [B,N,4]
    const float* anchors         = (const float*)d_in[2];  // [1,N,4]
    const float* annotations     = (const float*)d_in[3];  // [B,M,5]

    const int N = in_sizes[2] / 4;
    const int B = in_sizes[1] / (4 * N);
    const int C = in_sizes[0] / (B * N);
    const int M = in_sizes[3] / (5 * B);   // expected <= 64 (LDS tile)

    float*    acc   = (float*)d_ws;                           // 3*B floats
    uint32_t* state = (uint32_t*)((char*)d_ws + 256);         // B*N u32

    init_acc_kernel<<<1, 64, 0, stream>>>(acc, 3 * B);

    {
        const int blk = 320;                                  // 10 waves; N%320==0
        dim3 grid((N + blk - 1) / blk, B);
        assign_kernel<<<grid, blk, 0, stream>>>(anchors, regressions, annotations,
                                                state, acc, N, M, B);
    }
    {
        const size_t imgElems = (size_t)N * (size_t)C;
        const size_t quads = (imgElems + 3) / 4;
        const int blk = 256;
        dim3 grid((unsigned)((quads + blk - 1) / blk), B);
        cls_loss_kernel<<<grid, blk, 0, stream>>>(classifications, state, acc, N, C);
    }
    finalize_kernel<<<1, 32, 0, stream>>>(acc, (float*)d_out, B);
}